// ProjectiveLayer_37864431682255
// MI455X (gfx1250) — compile-verified
//
#include <hip/hip_runtime.h>
#include <hip/hip_bf16.h>
#include <stdint.h>

// Problem constants (match reference)
#define B_DIM   16
#define S_DIM   512
#define N_HASH  8
#define M_BLOOM 4096
#define W_WIN   2
#define K_BLK   (2 * W_WIN + 1)   // 5 shifted blocks

// -------------------------------------------------------------------------
// Kernel A: zero the 640 MB output using the CDNA5 async store-from-LDS path.
// Each lane streams 16 B of zeros from a shared 512 B LDS tile straight to
// global memory (no VGPR data operand; tracked on ASYNCcnt). NT store hint
// streams past the caches — nothing re-reads the bulk zeros except the tiny
// scatter in kernel B.
// -------------------------------------------------------------------------
__global__ void __launch_bounds__(256) bloom_zero_async(float4* __restrict__ out4,
                                                        unsigned long long n4) {
    __shared__ __align__(16) float4 zbuf[32];   // 512 B of zeros
    if (threadIdx.x < 32) {
        zbuf[threadIdx.x] = make_float4(0.0f, 0.0f, 0.0f, 0.0f);
    }
    __syncthreads();   // ds stores complete (compiler waits DScnt) before barrier

    // Flat LDS pointers carry the wave-relative LDS byte offset in addr[31:0];
    // the async-store VSRC operand wants exactly that offset.
    unsigned lds_off = (unsigned)(uintptr_t)(&zbuf[threadIdx.x & 31]);

    unsigned long long stride = (unsigned long long)gridDim.x * blockDim.x;
    for (unsigned long long i = (unsigned long long)blockIdx.x * blockDim.x + threadIdx.x;
         i < n4; i += stride) {
        unsigned long long gaddr = (unsigned long long)(out4 + i);
        // copy 16 B from LDS offset lds_off to global address gaddr, async (ASYNCcnt), non-temporal
        asm volatile("global_store_async_from_lds_b128 %0, %1, off th:TH_STORE_NT"
                     :: "v"(gaddr), "v"(lds_off)
                     : "memory");
    }
    // Drain all outstanding async stores before the wave retires.
    asm volatile("s_wait_asynccnt 0x0" ::: "memory");
}

// -------------------------------------------------------------------------
// Kernel B: sparse scatter. One thread per hash value (B*S*N = 65,536).
// Each hash h at (b, s) contributes +1 to bin m = h & 4095 of the bloom
// column s; the moving window replicates that into block k at output column
// s + (W - k). Collisions only occur among the 8 hashes of one source
// column, so plain f32 atomics (327,680 total) are essentially free.
// -------------------------------------------------------------------------
__global__ void __launch_bounds__(256) bloom_scatter(const int* __restrict__ hashes,
                                                     float* __restrict__ out) {
    int tid = blockIdx.x * blockDim.x + threadIdx.x;      // 0 .. B*S*N-1
    int s   = (tid / N_HASH) & (S_DIM - 1);
    int b   =  tid / (N_HASH * S_DIM);

    int h = hashes[tid];                                  // coalesced load
    int m = h & (M_BLOOM - 1);                            // hashes >= 0, M pow2

    const unsigned long long blkStride = (unsigned long long)M_BLOOM * S_DIM;
    unsigned long long base =
        (unsigned long long)b * (K_BLK * blkStride) + (unsigned long long)m * S_DIM;

#pragma unroll
    for (int k = 0; k < K_BLK; ++k) {
        int so = s + (W_WIN - k);                         // destination column
        if (0 <= so && so < S_DIM) {
            atomicAdd(&out[base + (unsigned long long)k * blkStride + (unsigned)so], 1.0f);
        }
    }
}

// -------------------------------------------------------------------------
// Host-side launch: zero-fill (bandwidth bound, ~27.5 us floor at 23.3 TB/s),
// then scatter (negligible).
// -------------------------------------------------------------------------
extern "C" void kernel_launch(void* const* d_in, const int* in_sizes, int n_in,
                              void* d_out, int out_size, void* d_ws, size_t ws_size,
                              hipStream_t stream) {
    const int* hashes = (const int*)d_in[0];     // [B, S, N] int32
    float* out = (float*)d_out;                  // [B, K_BLK*M, S] float32

    unsigned long long n4 = (unsigned long long)out_size / 4;  // float4 count (exact)

    // 8192 blocks x 256 threads: ~2M lanes, 20 async b128 stores each.
    bloom_zero_async<<<8192, 256, 0, stream>>>((float4*)out, n4);

    int total = B_DIM * S_DIM * N_HASH;          // 65,536
    bloom_scatter<<<total / 256, 256, 0, stream>>>(hashes, out);
}